// GCN_15865609192049
// MI455X (gfx1250) — compile-verified
//
#include <hip/hip_runtime.h>
#include <math.h>

#define Nn     50000
#define Ein    500000
#define Etot   (Ein + Nn)      // 550000 (with self-loops)
#define INCH   128
#define HID    64
#define NHEADS 4
#define C1     (NHEADS * HID)  // 256
#define OUTC   10
#define NG     64

typedef __attribute__((ext_vector_type(2))) float v2f;
typedef __attribute__((ext_vector_type(8))) float v8f;

// ---- monotone uint encoding of float for atomicMax-based segment_max ----
__device__ __forceinline__ unsigned enc_f32(float f) {
  unsigned u = __float_as_uint(f);
  return (u & 0x80000000u) ? ~u : (u | 0x80000000u);
}
__device__ __forceinline__ float dec_f32(unsigned u) {
  return __uint_as_float((u & 0x80000000u) ? (u ^ 0x80000000u) : ~u);
}

// =======================================================================
// GEMM1: h1[N,256] = x[N,128] @ W1[256,128]^T   (fp32 WMMA 16x16x4)
// one wave32 per 16x16 output tile; 32 k-steps of K=4
// =======================================================================
__global__ void gat_gemm1_wmma(const float* __restrict__ x,
                               const float* __restrict__ W1,
                               float* __restrict__ h1) {
  int wave = (int)((blockIdx.x * blockDim.x + threadIdx.x) >> 5);
  int lane = (int)(threadIdx.x & 31);
  const int nTilesM = Nn / 16;             // 3125 (exact)
  if (wave >= nTilesM * (C1 / 16)) return;
  int tm = wave >> 4;                      // M tile
  int tn = wave & 15;                      // N tile
  int mBase = tm * 16, nBase = tn * 16;
  int mr = mBase + (lane & 15);            // A row for this lane
  int nr = nBase + (lane & 15);            // B col for this lane
  int kOff = (lane < 16) ? 0 : 2;

  v8f acc = {0.f, 0.f, 0.f, 0.f, 0.f, 0.f, 0.f, 0.f};
  for (int kb = 0; kb < INCH; kb += 4) {
    v2f a, b;
    const float* ap = x  + (size_t)mr * INCH + kb + kOff;
    const float* bp = W1 + (size_t)nr * INCH + kb + kOff;
    a.x = ap[0]; a.y = ap[1];
    b.x = bp[0]; b.y = bp[1];
    acc = __builtin_amdgcn_wmma_f32_16x16x4_f32(false, a, false, b,
                                                (short)0, acc, false, false);
  }
  int col = nBase + (lane & 15);
  int rowBase = mBase + ((lane < 16) ? 0 : 8);
#pragma unroll
  for (int r = 0; r < 8; ++r)
    h1[(size_t)(rowBase + r) * C1 + col] = acc[r];
}

// =======================================================================
// GEMM2: h2[N,10] = h1act[N,256] @ W2[10,256]^T  (N padded to 16 w/ zeros)
// =======================================================================
__global__ void gat_gemm2_wmma(const float* __restrict__ h1act,
                               const float* __restrict__ W2,
                               float* __restrict__ h2) {
  int wave = (int)((blockIdx.x * blockDim.x + threadIdx.x) >> 5);
  int lane = (int)(threadIdx.x & 31);
  if (wave >= Nn / 16) return;
  int mBase = wave * 16;
  int n = lane & 15;
  int kOff = (lane < 16) ? 0 : 2;

  v8f acc = {0.f, 0.f, 0.f, 0.f, 0.f, 0.f, 0.f, 0.f};
  for (int kb = 0; kb < C1; kb += 4) {
    v2f a, b;
    const float* ap = h1act + (size_t)(mBase + n) * C1 + kb + kOff;
    a.x = ap[0]; a.y = ap[1];
    if (n < OUTC) {
      const float* bp = W2 + (size_t)n * C1 + kb + kOff;
      b.x = bp[0]; b.y = bp[1];
    } else {
      b.x = 0.f; b.y = 0.f;
    }
    acc = __builtin_amdgcn_wmma_f32_16x16x4_f32(false, a, false, b,
                                                (short)0, acc, false, false);
  }
  if (n < OUTC) {
    int rowBase = mBase + ((lane < 16) ? 0 : 8);
#pragma unroll
    for (int r = 0; r < 8; ++r)
      h2[(size_t)(rowBase + r) * OUTC + n] = acc[r];
  }
}

// ---- per-node attention coefficients, layer 1 --------------------------
__global__ void gat_attn1(const float* __restrict__ h1,
                          const float* __restrict__ asrc,
                          const float* __restrict__ adst,
                          float* __restrict__ as1, float* __restrict__ ad1) {
  int i = blockIdx.x * blockDim.x + threadIdx.x;   // over N*HEADS
  if (i >= Nn * NHEADS) return;
  int n = i >> 2, h = i & 3;
  const float* hp = h1 + (size_t)n * C1 + h * HID;
  const float* sp = asrc + h * HID;
  const float* dp = adst + h * HID;
  float ss = 0.f, dd = 0.f;
  for (int c = 0; c < HID; ++c) { float v = hp[c]; ss += v * sp[c]; dd += v * dp[c]; }
  as1[i] = ss; ad1[i] = dd;
}

__device__ __forceinline__ void edge_sd(int e, const int* ei, int& s, int& d) {
  if (e < Ein) { s = ei[e]; d = ei[Ein + e]; } else { s = e - Ein; d = s; }
}

// ---- layer-1 edge passes ----------------------------------------------
__global__ void gat_edge1_max(const int* __restrict__ ei,
                              const float* __restrict__ as1,
                              const float* __restrict__ ad1,
                              float* __restrict__ e1,
                              unsigned* __restrict__ emax1) {
  int e = blockIdx.x * blockDim.x + threadIdx.x;
  if (e >= Etot) return;
  int s, d; edge_sd(e, ei, s, d);
#pragma unroll
  for (int h = 0; h < NHEADS; ++h) {
    float v = as1[s * NHEADS + h] + ad1[d * NHEADS + h];
    v = (v > 0.f) ? v : 0.2f * v;                 // leaky_relu(., 0.2)
    e1[(size_t)e * NHEADS + h] = v;
    atomicMax(&emax1[d * NHEADS + h], enc_f32(v));
  }
}

__global__ void gat_edge1_sum(const int* __restrict__ ei,
                              float* __restrict__ e1,
                              const unsigned* __restrict__ emax1,
                              float* __restrict__ den1) {
  int e = blockIdx.x * blockDim.x + threadIdx.x;
  if (e >= Etot) return;
  int s, d; edge_sd(e, ei, s, d);
#pragma unroll
  for (int h = 0; h < NHEADS; ++h) {
    float ee = expf(e1[(size_t)e * NHEADS + h] - dec_f32(emax1[d * NHEADS + h]));
    e1[(size_t)e * NHEADS + h] = ee;
    atomicAdd(&den1[d * NHEADS + h], ee);
  }
}

__global__ void gat_edge1_agg(const int* __restrict__ ei,
                              const float* __restrict__ e1,
                              const float* __restrict__ den1,
                              const float* __restrict__ h1,
                              float* __restrict__ agg1) {
  int e = blockIdx.x;            // one block (256 thr) per edge
  int c = threadIdx.x;           // channel 0..255
  int s, d; edge_sd(e, ei, s, d);
  int h = c >> 6;
  float alpha = e1[(size_t)e * NHEADS + h] / (den1[d * NHEADS + h] + 1e-16f);
  atomicAdd(&agg1[(size_t)d * C1 + c], alpha * h1[(size_t)s * C1 + c]);
}

__global__ void gat_elu_bias(float* __restrict__ agg1, const float* __restrict__ b1) {
  int i = blockIdx.x * blockDim.x + threadIdx.x;
  if (i >= Nn * C1) return;
  float v = agg1[i] + b1[i & (C1 - 1)];
  agg1[i] = (v > 0.f) ? v : (expf(v) - 1.0f);     // elu
}

// ---- layer-2 ----------------------------------------------------------
__global__ void gat_attn2(const float* __restrict__ h2,
                          const float* __restrict__ asrc,
                          const float* __restrict__ adst,
                          float* __restrict__ as2, float* __restrict__ ad2) {
  int n = blockIdx.x * blockDim.x + threadIdx.x;
  if (n >= Nn) return;
  float ss = 0.f, dd = 0.f;
#pragma unroll
  for (int c = 0; c < OUTC; ++c) {
    float v = h2[(size_t)n * OUTC + c];
    ss += v * asrc[c]; dd += v * adst[c];
  }
  as2[n] = ss; ad2[n] = dd;
}

__global__ void gat_edge2_max(const int* __restrict__ ei,
                              const float* __restrict__ as2,
                              const float* __restrict__ ad2,
                              float* __restrict__ e2,
                              unsigned* __restrict__ emax2) {
  int e = blockIdx.x * blockDim.x + threadIdx.x;
  if (e >= Etot) return;
  int s, d; edge_sd(e, ei, s, d);
  float v = as2[s] + ad2[d];
  v = (v > 0.f) ? v : 0.2f * v;
  e2[e] = v;
  atomicMax(&emax2[d], enc_f32(v));
}

__global__ void gat_edge2_sum(const int* __restrict__ ei,
                              float* __restrict__ e2,
                              const unsigned* __restrict__ emax2,
                              float* __restrict__ den2) {
  int e = blockIdx.x * blockDim.x + threadIdx.x;
  if (e >= Etot) return;
  int s, d; edge_sd(e, ei, s, d);
  float ee = expf(e2[e] - dec_f32(emax2[d]));
  e2[e] = ee;
  atomicAdd(&den2[d], ee);
}

__global__ void gat_edge2_agg(const int* __restrict__ ei,
                              const float* __restrict__ e2,
                              const float* __restrict__ den2,
                              const float* __restrict__ h2,
                              float* __restrict__ agg2) {
  int e = blockIdx.x * blockDim.x + threadIdx.x;
  if (e >= Etot) return;
  int s, d; edge_sd(e, ei, s, d);
  float alpha = e2[e] / (den2[d] + 1e-16f);
#pragma unroll
  for (int c = 0; c < OUTC; ++c)
    atomicAdd(&agg2[(size_t)d * OUTC + c], alpha * h2[(size_t)s * OUTC + c]);
}

__global__ void gat_pool(const float* __restrict__ agg2,
                         const float* __restrict__ b2,
                         const int* __restrict__ batch,
                         float* __restrict__ pooled, float* __restrict__ counts) {
  int n = blockIdx.x * blockDim.x + threadIdx.x;
  if (n >= Nn) return;
  int g = batch[n];
  atomicAdd(&counts[g], 1.0f);
#pragma unroll
  for (int c = 0; c < OUTC; ++c)
    atomicAdd(&pooled[g * OUTC + c], agg2[(size_t)n * OUTC + c] + b2[c]);
}

__global__ void gat_logsoftmax(const float* __restrict__ pooled,
                               const float* __restrict__ counts,
                               float* __restrict__ out) {
  int g = threadIdx.x;
  if (g >= NG) return;
  float cnt = fmaxf(counts[g], 1.0f);
  float v[OUTC];
  float mx = -3.0e38f;
#pragma unroll
  for (int c = 0; c < OUTC; ++c) { v[c] = pooled[g * OUTC + c] / cnt; mx = fmaxf(mx, v[c]); }
  float sum = 0.f;
#pragma unroll
  for (int c = 0; c < OUTC; ++c) sum += expf(v[c] - mx);
  float lse = mx + logf(sum);
#pragma unroll
  for (int c = 0; c < OUTC; ++c) out[g * OUTC + c] = v[c] - lse;
}

// =======================================================================
extern "C" void kernel_launch(void* const* d_in, const int* in_sizes, int n_in,
                              void* d_out, int out_size, void* d_ws, size_t ws_size,
                              hipStream_t stream) {
  const float* x     = (const float*)d_in[0];
  const int*   ei    = (const int*)d_in[1];
  // d_in[2] edge_weight: ignored by GATConv (per reference)
  const int*   batch = (const int*)d_in[3];
  const float* W1    = (const float*)d_in[4];
  const float* asrc1 = (const float*)d_in[5];
  const float* adst1 = (const float*)d_in[6];
  const float* b1    = (const float*)d_in[7];
  const float* W2    = (const float*)d_in[8];
  const float* asrc2 = (const float*)d_in[9];
  const float* adst2 = (const float*)d_in[10];
  const float* b2    = (const float*)d_in[11];
  float* out = (float*)d_out;
  float* ws  = (float*)d_ws;

  // ---- workspace layout (float units) ----
  const size_t OFF_H1    = 0;                                    // N*C1
  const size_t OFF_AGG1  = OFF_H1   + (size_t)Nn * C1;           // N*C1
  const size_t OFF_AS1   = OFF_AGG1 + (size_t)Nn * C1;           // N*4
  const size_t OFF_AD1   = OFF_AS1  + (size_t)Nn * NHEADS;
  const size_t OFF_EMAX1 = OFF_AD1  + (size_t)Nn * NHEADS;       // u32 N*4
  const size_t OFF_DEN1  = OFF_EMAX1+ (size_t)Nn * NHEADS;
  const size_t OFF_E1    = OFF_DEN1 + (size_t)Nn * NHEADS;       // Etot*4
  // layer-2 scratch overlays the (dead after ELU) h1 region:
  const size_t OFF_H2    = OFF_H1;                               // N*OUTC
  const size_t OFF_AS2   = OFF_H2   + (size_t)Nn * OUTC;
  const size_t OFF_AD2   = OFF_AS2  + (size_t)Nn;
  const size_t OFF_EMAX2 = OFF_AD2  + (size_t)Nn;                // u32 N
  const size_t OFF_DEN2  = OFF_EMAX2+ (size_t)Nn;
  const size_t OFF_E2    = OFF_DEN2 + (size_t)Nn;                // Etot
  const size_t OFF_AGG2  = OFF_E2   + (size_t)Etot;              // N*OUTC
  const size_t OFF_POOL  = OFF_AGG2 + (size_t)Nn * OUTC;         // G*OUTC
  const size_t OFF_CNT   = OFF_POOL + (size_t)NG * OUTC;         // G

  float*    h1    = ws + OFF_H1;
  float*    agg1  = ws + OFF_AGG1;
  float*    as1   = ws + OFF_AS1;
  float*    ad1   = ws + OFF_AD1;
  unsigned* emax1 = (unsigned*)(ws + OFF_EMAX1);
  float*    den1  = ws + OFF_DEN1;
  float*    e1    = ws + OFF_E1;
  float*    h2    = ws + OFF_H2;
  float*    as2   = ws + OFF_AS2;
  float*    ad2   = ws + OFF_AD2;
  unsigned* emax2 = (unsigned*)(ws + OFF_EMAX2);
  float*    den2  = ws + OFF_DEN2;
  float*    e2    = ws + OFF_E2;
  float*    agg2  = ws + OFF_AGG2;
  float*    pooled= ws + OFF_POOL;
  float*    counts= ws + OFF_CNT;

  // ---- layer 1 ----
  hipMemsetAsync(agg1, 0, (size_t)Nn * C1 * sizeof(float), stream);
  hipMemsetAsync(emax1, 0, (size_t)Nn * NHEADS * 2 * sizeof(float), stream); // emax1+den1

  {
    int waves = (Nn / 16) * (C1 / 16);   // 50000
    int blocks = (waves * 32 + 255) / 256;
    gat_gemm1_wmma<<<blocks, 256, 0, stream>>>(x, W1, h1);
  }
  gat_attn1<<<(Nn * NHEADS + 255) / 256, 256, 0, stream>>>(h1, asrc1, adst1, as1, ad1);
  gat_edge1_max<<<(Etot + 255) / 256, 256, 0, stream>>>(ei, as1, ad1, e1, emax1);
  gat_edge1_sum<<<(Etot + 255) / 256, 256, 0, stream>>>(ei, e1, emax1, den1);
  gat_edge1_agg<<<Etot, 256, 0, stream>>>(ei, e1, den1, h1, agg1);
  gat_elu_bias<<<(Nn * C1 + 255) / 256, 256, 0, stream>>>(agg1, b1);   // agg1 -> h1act

  // ---- layer 2 (h1 region now dead; reused) ----
  hipMemsetAsync(emax2, 0, (size_t)Nn * 2 * sizeof(float), stream);                 // emax2+den2
  hipMemsetAsync(agg2, 0, ((size_t)Nn * OUTC + NG * OUTC + NG) * sizeof(float), stream); // agg2+pool+cnt

  {
    int waves = Nn / 16;                 // 3125
    int blocks = (waves * 32 + 255) / 256;
    gat_gemm2_wmma<<<blocks, 256, 0, stream>>>(agg1, W2, h2);
  }
  gat_attn2<<<(Nn + 255) / 256, 256, 0, stream>>>(h2, asrc2, adst2, as2, ad2);
  gat_edge2_max<<<(Etot + 255) / 256, 256, 0, stream>>>(ei, as2, ad2, e2, emax2);
  gat_edge2_sum<<<(Etot + 255) / 256, 256, 0, stream>>>(ei, e2, emax2, den2);
  gat_edge2_agg<<<(Etot + 255) / 256, 256, 0, stream>>>(ei, e2, den2, h2, agg2);

  gat_pool<<<(Nn + 255) / 256, 256, 0, stream>>>(agg2, b2, batch, pooled, counts);
  gat_logsoftmax<<<1, 64, 0, stream>>>(pooled, counts, out);

  (void)in_sizes; (void)n_in; (void)out_size; (void)ws_size;
}